// MLPAttention_74062416052582
// MI455X (gfx1250) — compile-verified
//
#include <hip/hip_runtime.h>

// ---------------------------------------------------------------------------
// MLPAttention for MI455X (gfx1250, wave32).
//   h1 = query @ Wq^T + bq            [B,Q,128]   (WMMA f32 16x16x4)
//   h2 = key   @ Wk^T + bk            [B,K,128]   (WMMA f32 16x16x4)
//   logits[b,q,k] = sum_a relu(h1+h2)*Wa + ba     (fused, VALU, never
//                                                  materializes [B,Q,K,A])
//   out = softmax_k(logits) @ value   [B,Q,512]   (flash-style online softmax
//                                                  + WMMA f32 16x16x4 P@V)
// ---------------------------------------------------------------------------

typedef __attribute__((ext_vector_type(2))) float v2f;
typedef __attribute__((ext_vector_type(8))) float v8f;
typedef __attribute__((ext_vector_type(4))) int   v4i;

#define ADIM 128   // ATTN_SIZE
#define DHID 512   // KEY_SIZE == QUERY_SIZE
#define DDIM 512   // value feature dim
#define QLEN 256
#define KLEN 1024
#define NB   4
#define KC   128   // flash K-chunk

// ---- gfx1250 async memory->LDS path (guarded: falls back to VGPR staging) --
#if __has_builtin(__builtin_amdgcn_global_load_async_to_lds_b128) && \
    __has_builtin(__builtin_amdgcn_s_wait_asynccnt)
#define HAVE_ASYNC_LDS 1
typedef __attribute__((address_space(1))) v4i gv4i_t;   // global int4
typedef __attribute__((address_space(3))) v4i lv4i_t;   // LDS int4
__device__ __forceinline__ void async_copy16(const void* g, void* l) {
  // global_load_async_to_lds_b128 (tracked by ASYNCcnt)
  __builtin_amdgcn_global_load_async_to_lds_b128(
      (gv4i_t*)(v4i*)const_cast<void*>(g),
      (lv4i_t*)(v4i*)l,
      /*imm_offset=*/0, /*cpol=*/0);
}
#else
#define HAVE_ASYNC_LDS 0
#endif

__device__ __forceinline__ v8f wmma4(v2f a, v2f b, v8f c) {
  // D(16x16,f32) = A(16x4,f32) * B(4x16,f32) + C
  return __builtin_amdgcn_wmma_f32_16x16x4_f32(
      /*neg_a=*/false, a, /*neg_b=*/false, b,
      /*c_mod=*/(short)0, c, /*reuse_a=*/false, /*reuse_b=*/false);
}

// ---------------------------------------------------------------------------
// H[m, n0:n0+16] = X[m, :] @ W[n, :]^T + bias[n]   (one 16x16 tile per wave)
// A layout (16x4 f32): lanes 0-15 carry M=0..15 with K=k,k+1; lanes 16-31 K=k+2,k+3
// B layout (4x16 f32): lanes 0-15 carry N=0..15 with K=k,k+1; lanes 16-31 K=k+2,k+3
// ---------------------------------------------------------------------------
__global__ __launch_bounds__(128) void proj_kernel(
    const float* __restrict__ X, const float* __restrict__ W,
    const float* __restrict__ bias, float* __restrict__ H) {
  const int wave = threadIdx.x >> 5;
  const int lane = threadIdx.x & 31;
  const int half = lane >> 4;
  const int l16  = lane & 15;
  const int m0 = (blockIdx.x * 4 + wave) * 16;
  const int n0 = blockIdx.y * 16;

  const float* xrow = X + (size_t)(m0 + l16) * DHID;
  const float* wrow = W + (size_t)(n0 + l16) * DHID;

  v8f acc = {};
  #pragma unroll 4
  for (int k = 0; k < DHID; k += 4) {
    const int ka = k + half * 2;
    v2f a = { xrow[ka], xrow[ka + 1] };
    v2f b = { wrow[ka], wrow[ka + 1] };
    acc = wmma4(a, b, acc);
  }

  const float bn = bias[n0 + l16];
  // C/D layout: VGPR i -> row m0+i (lanes 0-15) / m0+8+i (lanes 16-31)
  float* orow = H + (size_t)(m0 + half * 8) * ADIM + n0 + l16;
  #pragma unroll
  for (int i = 0; i < 8; ++i)
    orow[(size_t)i * ADIM] = acc[i] + bn;
}

// ---------------------------------------------------------------------------
// Fused MLP-attention: one block per (batch, 16-query tile), 256 threads
// (8 waves). Flash loop over K in chunks of KC.
// ---------------------------------------------------------------------------
__global__ __launch_bounds__(256) void attn_kernel(
    const float* __restrict__ H1, const float* __restrict__ H2,
    const float* __restrict__ Wa, const float* __restrict__ ba,
    const float* __restrict__ V,  float* __restrict__ O) {
  __shared__ __align__(16) float s_h1[16][ADIM + 4];  // query-side hidden tile
  __shared__ float s_p[16][KC + 4];                   // logits / probs chunk
  __shared__ float s_wa[ADIM];
  __shared__ float s_m[16], s_s[16], s_scale[16], s_inv[16];

  const int b    = blockIdx.x;
  const int q0   = blockIdx.y * 16;
  const int tid  = threadIdx.x;
  const int wave = tid >> 5;
  const int lane = tid & 31;
  const int half = lane >> 4;
  const int l16  = lane & 15;

  // ---- stage h1 tile (async memory->LDS when available), Wa, softmax state
  {
    const float* h1 = H1 + ((size_t)b * QLEN + q0) * ADIM;
#if HAVE_ASYNC_LDS
    // 16 rows x 128 floats = 512 x 16B packets; 2 per thread, straight to LDS.
    for (int i = tid; i < 16 * (ADIM / 4); i += 256) {
      const int row  = i >> 5;        // ADIM/4 == 32 packets per row
      const int col4 = (i & 31) * 4;
      async_copy16(h1 + (size_t)row * ADIM + col4, &s_h1[row][col4]);
    }
    __builtin_amdgcn_s_wait_asynccnt(0);
#else
    for (int i = tid; i < 16 * ADIM; i += 256)
      s_h1[i >> 7][i & (ADIM - 1)] = h1[i];
#endif
    if (tid < ADIM) s_wa[tid] = Wa[tid];
    if (tid < 16) { s_m[tid] = -3.0e38f; s_s[tid] = 0.0f; }
  }
  __syncthreads();

  const float bav = ba[0];
  const int kq = tid >> 1;        // k within chunk (two threads per k)
  const int qb = (tid & 1) * 8;   // each handles 8 of the 16 query rows
  const int r  = tid >> 4;        // softmax: row
  const int cg = tid & 15;        // softmax: 16 lanes per row

  v8f acc0 = {}, acc1 = {}, acc2 = {}, acc3 = {};
  const float* vb = V + (size_t)b * KLEN * DDIM;

  for (int c = 0; c < KLEN / KC; ++c) {
    // ---- chunk logits: s_p[q][k] = sum_a relu(h1[q][a]+h2[k][a])*Wa[a] + ba
    {
      const float* h2 = H2 + ((size_t)b * KLEN + c * KC + kq) * ADIM;
      if (c + 1 < KLEN / KC)
        __builtin_prefetch(h2 + (size_t)KC * ADIM, 0, 0);  // global_prefetch_b8
      float la[8];
      #pragma unroll
      for (int q = 0; q < 8; ++q) la[q] = 0.0f;
      #pragma unroll 4
      for (int a = 0; a < ADIM; ++a) {
        const float h2a = h2[a];
        const float wa  = s_wa[a];
        #pragma unroll
        for (int q = 0; q < 8; ++q) {
          const float v = fmaxf(s_h1[qb + q][a] + h2a, 0.0f);
          la[q] = fmaf(v, wa, la[q]);
        }
      }
      #pragma unroll
      for (int q = 0; q < 8; ++q) s_p[qb + q][kq] = la[q] + bav;
    }
    __syncthreads();

    // ---- online softmax update for this chunk (16 lanes per row; xor-shuffles
    //      with mask < 16 stay inside each 16-lane half of the wave32)
    {
      const float mold = s_m[r];
      float mc = mold;
      #pragma unroll
      for (int k = cg; k < KC; k += 16) mc = fmaxf(mc, s_p[r][k]);
      #pragma unroll
      for (int off = 8; off >= 1; off >>= 1)
        mc = fmaxf(mc, __shfl_xor(mc, off, 32));
      float cs = 0.0f;
      #pragma unroll
      for (int k = cg; k < KC; k += 16) {
        const float e = __expf(s_p[r][k] - mc);
        s_p[r][k] = e;
        cs += e;
      }
      #pragma unroll
      for (int off = 8; off >= 1; off >>= 1) cs += __shfl_xor(cs, off, 32);
      if (cg == 0) {
        const float sc = __expf(mold - mc);
        s_scale[r] = sc;
        s_s[r] = s_s[r] * sc + cs;
        s_m[r] = mc;
      }
    }
    __syncthreads();

    // ---- rescale accumulators, then P(16xKC) @ V(KCx512) with WMMA.
    //      Each wave owns N-tiles {wave, wave+8, wave+16, wave+24} * 16.
    {
      float sc[8];
      #pragma unroll
      for (int i = 0; i < 8; ++i) sc[i] = s_scale[i + half * 8];
      #pragma unroll
      for (int i = 0; i < 8; ++i) {
        acc0[i] *= sc[i]; acc1[i] *= sc[i];
        acc2[i] *= sc[i]; acc3[i] *= sc[i];
      }
      const float* vc = vb + (size_t)(c * KC) * DDIM;
      for (int k = 0; k < KC; k += 4) {
        const int ka = k + half * 2;
        v2f a = { s_p[l16][ka], s_p[l16][ka + 1] };      // A: P rows
        const float* vk0 = vc + (size_t)ka * DDIM + l16; // B: V rows
        const float* vk1 = vk0 + DDIM;
        v2f b0 = { vk0[(wave)      * 16], vk1[(wave)      * 16] };
        v2f b1 = { vk0[(wave +  8) * 16], vk1[(wave +  8) * 16] };
        v2f b2 = { vk0[(wave + 16) * 16], vk1[(wave + 16) * 16] };
        v2f b3 = { vk0[(wave + 24) * 16], vk1[(wave + 24) * 16] };
        acc0 = wmma4(a, b0, acc0);
        acc1 = wmma4(a, b1, acc1);
        acc2 = wmma4(a, b2, acc2);
        acc3 = wmma4(a, b3, acc3);
      }
    }
    __syncthreads();
  }

  if (tid < 16) s_inv[tid] = 1.0f / s_s[tid];
  __syncthreads();

  float inv[8];
  #pragma unroll
  for (int i = 0; i < 8; ++i) inv[i] = s_inv[i + half * 8];
  float* ob = O + ((size_t)b * QLEN + q0 + half * 8) * DDIM + l16;
  #pragma unroll
  for (int i = 0; i < 8; ++i) {
    ob[(size_t)i * DDIM + (wave)      * 16] = acc0[i] * inv[i];
    ob[(size_t)i * DDIM + (wave +  8) * 16] = acc1[i] * inv[i];
    ob[(size_t)i * DDIM + (wave + 16) * 16] = acc2[i] * inv[i];
    ob[(size_t)i * DDIM + (wave + 24) * 16] = acc3[i] * inv[i];
  }
}

// ---------------------------------------------------------------------------
extern "C" void kernel_launch(void* const* d_in, const int* in_sizes, int n_in,
                              void* d_out, int out_size, void* d_ws, size_t ws_size,
                              hipStream_t stream) {
  const float* key   = (const float*)d_in[0];  // [B,K,512]
  const float* query = (const float*)d_in[1];  // [B,Q,512]
  const float* value = (const float*)d_in[2];  // [B,K,512]
  const float* Wk    = (const float*)d_in[3];  // [128,512]
  const float* bk    = (const float*)d_in[4];  // [128]
  const float* Wq    = (const float*)d_in[5];  // [128,512]
  const float* bq    = (const float*)d_in[6];  // [128]
  const float* Wa    = (const float*)d_in[7];  // [128]
  const float* ba    = (const float*)d_in[8];  // [1]
  float* out = (float*)d_out;                  // [B,Q,512]

  float* h1 = (float*)d_ws;                        // [B*Q, 128] = 512 KB
  float* h2 = h1 + (size_t)NB * QLEN * ADIM;       // [B*K, 128] = 2 MB

  // h1 = query @ Wq^T + bq
  proj_kernel<<<dim3((NB * QLEN) / 64, ADIM / 16), 128, 0, stream>>>(query, Wq, bq, h1);
  // h2 = key @ Wk^T + bk
  proj_kernel<<<dim3((NB * KLEN) / 64, ADIM / 16), 128, 0, stream>>>(key, Wk, bk, h2);
  // fused scoring + softmax + score@value
  attn_kernel<<<dim3(NB, QLEN / 16), 256, 0, stream>>>(h1, h2, Wa, ba, value, out);
}